// TolmanEichenbaumMachine_37838661878393
// MI455X (gfx1250) — compile-verified
//
#include <hip/hip_runtime.h>
#include <hip/hip_bf16.h>

// ---------------- problem constants ----------------
#define BSZ   256
#define TT    8
#define PDIM  400
#define GDIM  120
#define NOBJ  45
#define SDIM  10
#define HIDD  200
#define GDOWN 40
#define XFLAT 50

typedef float v2f __attribute__((ext_vector_type(2)));
typedef float v8f __attribute__((ext_vector_type(8)));

__device__ __forceinline__ float leaky_clamp(float x) {
    float t = fminf(fmaxf(x, -1.0f), 1.0f);
    return t > 0.0f ? t : 0.01f * t;
}
__device__ __forceinline__ float scale_log_sigma(float r) {
    return -2.0f + 6.0f * tanhf(r * (1.0f / 6.0f));
}

// ---------------- workspace layout (floats) ----------------
// M: [256][400][400], x_filt: [256][50], then 7 [256][400] arrays.
static constexpr size_t OFF_M    = 0;
static constexpr size_t SZ_M     = (size_t)BSZ * PDIM * PDIM;   // 40,960,000
static constexpr size_t OFF_XF   = OFF_M + SZ_M;
static constexpr size_t SZ_XF    = (size_t)BSZ * XFLAT;         // 12,800
static constexpr size_t OFF_PINF = OFF_XF + SZ_XF;
static constexpr size_t SZ_BP    = (size_t)BSZ * PDIM;          // 102,400
static constexpr size_t OFF_H    = OFF_PINF + SZ_BP;
static constexpr size_t OFF_Z1   = OFF_H + SZ_BP;
static constexpr size_t OFF_Z2   = OFF_Z1 + SZ_BP;
static constexpr size_t OFF_PT   = OFF_Z2 + SZ_BP;
static constexpr size_t OFF_U    = OFF_PT + SZ_BP;
static constexpr size_t OFF_V    = OFF_U + SZ_BP;

// ---------------- kernels ----------------

__global__ __launch_bounds__(256) void k_zero(float4* __restrict__ p, long n4) {
    long i = (long)blockIdx.x * blockDim.x + threadIdx.x;
    if (i < n4) p[i] = make_float4(0.f, 0.f, 0.f, 0.f);
}

// Per-batch prep: sensory filter update, g-downsample, g_exp/x_exp skinny GEMVs,
// p_inf and attractor seed h. One block per batch element.
__global__ __launch_bounds__(64) void k_prep(
    const float* __restrict__ g_seq, const int* __restrict__ x_idx,
    const float* __restrict__ W_comp, const float* __restrict__ W_repeat,
    const float* __restrict__ W_tile,
    float* __restrict__ x_filt, float* __restrict__ p_inf, float* __restrict__ h,
    int t)
{
    __shared__ float xf[XFLAT];
    __shared__ float gd[GDOWN];
    const int b = blockIdx.x, tid = threadIdx.x;
    const float freqs[5] = {0.01f, 0.7f, 0.91f, 0.97f, 0.99f};
    const int obj = x_idx[b * TT + t];

    if (tid < XFLAT) {                       // x_c = W_comp[obj]; EMA filter per stream
        int s = tid / SDIM, d = tid % SDIM;
        float xc = W_comp[obj * SDIM + d];
        float f  = freqs[s];
        float v  = (1.0f - f) * x_filt[b * XFLAT + tid] + f * xc;
        x_filt[b * XFLAT + tid] = v;
        xf[tid] = v;
    }
    if (tid < GDOWN) {                       // downsample: first g_down dims per stream
        int g;
        if      (tid < 10) g = tid;
        else if (tid < 20) g = 30  + (tid - 10);
        else if (tid < 28) g = 60  + (tid - 20);
        else if (tid < 34) g = 84  + (tid - 28);
        else               g = 102 + (tid - 34);
        gd[tid] = g_seq[(b * TT + t) * GDIM + g];
    }
    __syncthreads();

    for (int p = tid; p < PDIM; p += 64) {
        float ge = 0.f;
        #pragma unroll
        for (int k = 0; k < GDOWN; ++k) ge += gd[k] * W_repeat[p * GDOWN + k];
        float xe = 0.f;
        #pragma unroll
        for (int k = 0; k < XFLAT; ++k) xe += xf[k] * W_tile[p * XFLAT + k];
        p_inf[b * PDIM + p] = leaky_clamp(ge * xe);
        h[b * PDIM + p]     = leaky_clamp(ge);       // attractor seed
    }
}

// Attractor retrieval: 5 iters of h = leaky_clamp(0.8*h + M[b] @ h).
// One block (8 waves) per batch element; h double-buffered in LDS; split-K across
// lanes with wave32 shfl_xor reduction; M streamed from L2 with float4 + prefetch.
__global__ __launch_bounds__(256) void k_attractor(
    const float* __restrict__ M, float* __restrict__ h)
{
    __shared__ float4 bufA[PDIM / 4], bufB[PDIM / 4];
    float* hA = (float*)bufA;
    float* hB = (float*)bufB;
    const int b = blockIdx.x, tid = threadIdx.x;
    const int w = tid >> 5, l = tid & 31;
    const float* Mb = M + (size_t)b * PDIM * PDIM;

    for (int k = tid; k < PDIM; k += 256) hA[k] = h[b * PDIM + k];
    __syncthreads();

    float* cur = hA;
    float* nxt = hB;
    for (int it = 0; it < 5; ++it) {
        const float4* cur4 = (const float4*)cur;
        for (int r = 0; r < 50; ++r) {               // wave w owns rows [w*50, w*50+50)
            const int i = w * 50 + r;
            const float4* mrow4 = (const float4*)(Mb + (size_t)i * PDIM);
            __builtin_prefetch((const void*)(mrow4 + 100 + l), 0, 3); // next row
            float sum = 0.f;
            #pragma unroll
            for (int c = 0; c < 3; ++c) {
                float4 m4 = mrow4[l + 32 * c];
                float4 h4 = cur4[l + 32 * c];
                sum += m4.x * h4.x + m4.y * h4.y + m4.z * h4.z + m4.w * h4.w;
            }
            if (l < 4) {                             // tail: 400 = (3*32+4)*4
                float4 m4 = mrow4[96 + l];
                float4 h4 = cur4[96 + l];
                sum += m4.x * h4.x + m4.y * h4.y + m4.z * h4.z + m4.w * h4.w;
            }
            #pragma unroll
            for (int off = 16; off > 0; off >>= 1)
                sum += __shfl_xor(sum, off, 32);
            if (l == 0) nxt[i] = leaky_clamp(0.8f * cur[i] + sum);
        }
        __syncthreads();
        float* tmp = cur; cur = nxt; nxt = tmp;
    }
    for (int k = tid; k < PDIM; k += 256) h[b * PDIM + k] = cur[k];
}

// WMMA GEMMs: Z1 = p_inf @ W_lv1 ; Z2 = p_ret @ W_lv2  ([256x400]x[400x400] f32).
// One wave32 per 16x16 output tile; K-loop of V_WMMA_F32_16X16X4_F32.
// A 16x4 layout: lane L holds row (rb + L%16), k = kb+2*(L/16) in v0, +1 in v1.
// B 4x16 layout: v0 = W[kb+2*(L/16)][nb+L%16], v1 = next-k row.
// 800 tiles total = 200 blocks * 4 waves exactly (EXEC all-ones, uniform flow).
__global__ __launch_bounds__(128) void k_gemm_lv(
    const float* __restrict__ p_inf, const float* __restrict__ p_ret,
    const float* __restrict__ W_lv1, const float* __restrict__ W_lv2,
    float* __restrict__ Z1, float* __restrict__ Z2)
{
    const int wid   = (blockIdx.x * 128 + threadIdx.x) >> 5;   // 0..799
    const int lane  = threadIdx.x & 31;
    const int which = wid / 400;
    const int tile  = wid % 400;
    const int mt = tile / 25, nt = tile % 25;                  // 16 x 25 tiles
    const float* A = which ? p_ret : p_inf;
    const float* W = which ? W_lv2 : W_lv1;
    float*       Z = which ? Z2 : Z1;
    const int rb = mt << 4, nb = nt << 4;
    const int lo = lane & 15, hi = lane >> 4;
    const float* arow = A + (rb + lo) * PDIM;
    const float* wcol = W + nb + lo;

    v8f c = {0.f, 0.f, 0.f, 0.f, 0.f, 0.f, 0.f, 0.f};
    #pragma unroll 4
    for (int kb = 0; kb < PDIM; kb += 4) {
        v2f a = *(const v2f*)(arow + kb + 2 * hi);
        v2f bm;
        bm.x = wcol[(kb + 2 * hi) * PDIM];
        bm.y = wcol[(kb + 2 * hi + 1) * PDIM];
        c = __builtin_amdgcn_wmma_f32_16x16x4_f32(false, a, false, bm,
                                                  (short)0, c, false, false);
    }
    // C/D 16x16 f32 layout: VGPR r, lanes0-15 -> M=r, lanes16-31 -> M=r+8
    #pragma unroll
    for (int r = 0; r < 8; ++r)
        Z[(rb + r + 8 * hi) * PDIM + nb + lo] = c[r];
}

// Precision-weighted fusion: lv = scale_log_sigma(Z + b); Gaussian fuse; emit
// p_t and the Hebbian rank-1 factors u = p_t - p_ret, v = p_t + p_ret.
__global__ __launch_bounds__(256) void k_fuse(
    const float* __restrict__ Z1, const float* __restrict__ Z2,
    const float* __restrict__ b_lv1, const float* __restrict__ b_lv2,
    const float* __restrict__ p_inf, const float* __restrict__ p_ret,
    float* __restrict__ p_t, float* __restrict__ u, float* __restrict__ v)
{
    const int idx = blockIdx.x * 256 + threadIdx.x;
    if (idx >= BSZ * PDIM) return;
    const int p = idx % PDIM;
    float lv1 = scale_log_sigma(Z1[idx] + b_lv1[p]);
    float lv2 = scale_log_sigma(Z2[idx] + b_lv2[p]);
    float i1 = expf(-lv1), i2 = expf(-lv2);
    float vc = 1.0f / (i1 + i2);
    float pr = p_ret[idx];
    float pt = vc * (p_inf[idx] * i1 + pr * i2);
    p_t[idx] = pt;
    u[idx]   = pt - pr;
    v[idx]   = pt + pr;
}

// Hebbian RMW: M[b] = 0.9999*M[b] + 0.5 * u v^T. One block per b, float4 streams.
__global__ __launch_bounds__(256) void k_hebb(
    float* __restrict__ M, const float* __restrict__ u, const float* __restrict__ v)
{
    __shared__ float  us[PDIM];
    __shared__ float4 vs4[PDIM / 4];
    const int b = blockIdx.x, tid = threadIdx.x;
    for (int k = tid; k < PDIM; k += 256) {
        us[k] = u[b * PDIM + k];
        ((float*)vs4)[k] = v[b * PDIM + k];
    }
    __syncthreads();
    float4* Mb4 = (float4*)(M + (size_t)b * PDIM * PDIM);
    for (int e = tid; e < (PDIM * PDIM) / 4; e += 256) {
        int i  = e / (PDIM / 4);
        int j4 = e % (PDIM / 4);
        float4 m  = Mb4[e];
        float4 vv = vs4[j4];
        float  s  = 0.5f * us[i];
        m.x = 0.9999f * m.x + s * vv.x;
        m.y = 0.9999f * m.y + s * vv.y;
        m.z = 0.9999f * m.z + s * vv.z;
        m.w = 0.9999f * m.w + s * vv.w;
        Mb4[e] = m;
    }
}

// Generative decode: x_s = w_x*(p_t[:, :100] @ W_tile0) + b_x;
// logits = elu(x_s @ W1 + b1) @ W2 + b2. One block per batch element.
__global__ __launch_bounds__(64) void k_decode(
    const float* __restrict__ p_t, const float* __restrict__ W_tile0,
    const float* __restrict__ w_x, const float* __restrict__ b_x,
    const float* __restrict__ W1, const float* __restrict__ b1,
    const float* __restrict__ W2, const float* __restrict__ b2,
    float* __restrict__ out, int t)
{
    __shared__ float p100[100];
    __shared__ float xs[SDIM];
    __shared__ float hm[HIDD];
    const int b = blockIdx.x, tid = threadIdx.x;
    for (int k = tid; k < 100; k += 64) p100[k] = p_t[b * PDIM + k];
    __syncthreads();
    if (tid < SDIM) {
        float a = 0.f;
        for (int k = 0; k < 100; ++k) a += p100[k] * W_tile0[k * SDIM + tid];
        xs[tid] = w_x[0] * a + b_x[tid];
    }
    __syncthreads();
    for (int j = tid; j < HIDD; j += 64) {
        float a = b1[j];
        #pragma unroll
        for (int d = 0; d < SDIM; ++d) a += xs[d] * W1[d * HIDD + j];
        hm[j] = a > 0.f ? a : expm1f(a);             // ELU
    }
    __syncthreads();
    if (tid < NOBJ) {
        float a = b2[tid];
        for (int j = 0; j < HIDD; ++j) a += hm[j] * W2[j * NOBJ + tid];
        out[(b * TT + t) * NOBJ + tid] = a;
    }
}

// ---------------- host launcher ----------------
extern "C" void kernel_launch(void* const* d_in, const int* in_sizes, int n_in,
                              void* d_out, int out_size, void* d_ws, size_t ws_size,
                              hipStream_t stream) {
    (void)in_sizes; (void)n_in; (void)out_size; (void)ws_size;
    const float* g_seq    = (const float*)d_in[0];
    const int*   x_idx    = (const int*)  d_in[1];
    const float* W_comp   = (const float*)d_in[2];
    const float* W_repeat = (const float*)d_in[3];
    const float* W_tile   = (const float*)d_in[4];
    const float* W_tile0  = (const float*)d_in[5];
    const float* w_x      = (const float*)d_in[6];
    const float* b_x      = (const float*)d_in[7];
    const float* W1       = (const float*)d_in[8];
    const float* b1       = (const float*)d_in[9];
    const float* W2       = (const float*)d_in[10];
    const float* b2       = (const float*)d_in[11];
    const float* W_lv1    = (const float*)d_in[12];
    const float* b_lv1    = (const float*)d_in[13];
    const float* W_lv2    = (const float*)d_in[14];
    const float* b_lv2    = (const float*)d_in[15];
    float* out = (float*)d_out;
    float* ws  = (float*)d_ws;

    float* M     = ws + OFF_M;
    float* xfilt = ws + OFF_XF;
    float* p_inf = ws + OFF_PINF;
    float* h     = ws + OFF_H;     // seed -> p_ret
    float* Z1    = ws + OFF_Z1;
    float* Z2    = ws + OFF_Z2;
    float* p_t   = ws + OFF_PT;
    float* u     = ws + OFF_U;
    float* v     = ws + OFF_V;

    // Zero Hebbian memory + sensory filters (contiguous region).
    const long n4 = (long)((SZ_M + SZ_XF) / 4);
    k_zero<<<(unsigned)((n4 + 255) / 256), 256, 0, stream>>>((float4*)ws, n4);

    for (int t = 0; t < TT; ++t) {
        k_prep<<<BSZ, 64, 0, stream>>>(g_seq, x_idx, W_comp, W_repeat, W_tile,
                                       xfilt, p_inf, h, t);
        k_attractor<<<BSZ, 256, 0, stream>>>(M, h);
        k_gemm_lv<<<200, 128, 0, stream>>>(p_inf, h, W_lv1, W_lv2, Z1, Z2);
        k_fuse<<<(BSZ * PDIM) / 256, 256, 0, stream>>>(Z1, Z2, b_lv1, b_lv2,
                                                       p_inf, h, p_t, u, v);
        k_hebb<<<BSZ, 256, 0, stream>>>(M, u, v);
        k_decode<<<BSZ, 64, 0, stream>>>(p_t, W_tile0, w_x, b_x, W1, b1, W2, b2,
                                         out, t);
    }
}